// BaseLL_31490700215028
// MI455X (gfx1250) — compile-verified
//
#include <hip/hip_runtime.h>

#define ZD 32
#define YD 16
#define KMIX 8
#define HIDN 64
#define TT 128
#define BB 512
#define JITF 1e-5f
#define LOG2PI 1.8378770664093453f

typedef float v2f __attribute__((ext_vector_type(2)));
typedef float v8f __attribute__((ext_vector_type(8)));

// D(32x32) = A(32x32) * B(32x32)   (or * B^T if transB), all row-major LDS.
// Uses V_WMMA_F32_16X16X4_F32: A frag 16x4 (lanes 0-15: K=0,1 ; lanes 16-31: K=2,3),
// B frag 4x16 row-striped, C/D 16x16 (VGPR r: M=r / M=r+8).
__device__ __forceinline__ void mm32(float* __restrict__ D, const float* __restrict__ A,
                                     const float* __restrict__ B, bool transB, int lane) {
  const int half = lane >> 4;
  const int lo = lane & 15;
  #pragma unroll
  for (int mi = 0; mi < 32; mi += 16) {
    #pragma unroll
    for (int ni = 0; ni < 32; ni += 16) {
      v8f c = {0.f, 0.f, 0.f, 0.f, 0.f, 0.f, 0.f, 0.f};
      #pragma unroll
      for (int k0 = 0; k0 < 32; k0 += 4) {
        const int ka = k0 + 2 * half;
        v2f a, b;
        a.x = A[(mi + lo) * 32 + ka];
        a.y = A[(mi + lo) * 32 + ka + 1];
        if (transB) {
          b.x = B[(ni + lo) * 32 + ka];
          b.y = B[(ni + lo) * 32 + ka + 1];
        } else {
          b.x = B[ka * 32 + (ni + lo)];
          b.y = B[(ka + 1) * 32 + (ni + lo)];
        }
        c = __builtin_amdgcn_wmma_f32_16x16x4_f32(false, a, false, b, (short)0, c, false, false);
      }
      #pragma unroll
      for (int r = 0; r < 8; ++r)
        D[(mi + r + 8 * half) * 32 + (ni + lo)] = c[r];
    }
  }
}

// In-place lower Cholesky of 32x32 LDS matrix, lane = row.
__device__ __forceinline__ void chol32(float* M, int lane) {
  for (int k = 0; k < ZD; ++k) {
    if (lane == k) M[k * ZD + k] = sqrtf(M[k * ZD + k]);
    __syncthreads();
    float dk = M[k * ZD + k];
    if (lane > k) M[lane * ZD + k] /= dk;
    __syncthreads();
    if (lane > k) {
      float lik = M[lane * ZD + k];
      for (int j = k + 1; j <= lane; ++j)
        M[lane * ZD + j] -= lik * M[j * ZD + k];
    }
    __syncthreads();
  }
}

__global__ __launch_bounds__(32) void kf_scan_kernel(
    const float* __restrict__ y, const float* __restrict__ mask, const float* __restrict__ times,
    const float* __restrict__ mu0, const float* __restrict__ S0d, const float* __restrict__ Hg,
    const float* __restrict__ Rdg, const float* __restrict__ Ab, const float* __restrict__ W1g,
    const float* __restrict__ b1g, const float* __restrict__ W2g, const float* __restrict__ b2g,
    const float* __restrict__ Qdg, const int* __restrict__ nsubp,
    float* __restrict__ out_mus, float* __restrict__ out_Ls, float* __restrict__ out_lp) {
  __shared__ float sH[YD * ZD];
  __shared__ float sW1[ZD * HIDN];
  __shared__ float sW2[HIDN * KMIX];
  __shared__ float sb1[HIDN];
  __shared__ float sb2[KMIX];
  __shared__ float sRd[YD];
  __shared__ float sQd[ZD];
  __shared__ float Lm[ZD * ZD];
  __shared__ float Sg[ZD * ZD];
  __shared__ float T1[ZD * ZD];
  __shared__ float T2[ZD * ZD];
  __shared__ float Am[ZD * ZD];
  __shared__ float Ssm[YD * YD];
  __shared__ float HPm[YD * ZD];
  __shared__ float Xs[YD * ZD];
  __shared__ float muv[ZD];
  __shared__ float muu[ZD];
  __shared__ float innov[YD];
  __shared__ float hidv[HIDN];
  __shared__ float alph[KMIX];
  __shared__ float lgv[KMIX];

  const int lane = threadIdx.x;
  const int b = blockIdx.x;
  const int nsub = *nsubp;
  const float dt = times[1] - times[0];
  const float h = dt / (float)nsub;

  for (int e = lane; e < YD * ZD; e += 32) sH[e] = Hg[e];
  for (int e = lane; e < ZD * HIDN; e += 32) sW1[e] = W1g[e];
  for (int e = lane; e < HIDN * KMIX; e += 32) sW2[e] = W2g[e];
  for (int e = lane; e < HIDN; e += 32) sb1[e] = b1g[e];
  if (lane < KMIX) sb2[lane] = b2g[lane];
  if (lane < YD) sRd[lane] = Rdg[lane];
  sQd[lane] = Qdg[lane];
  muv[lane] = mu0[lane];
  for (int j = 0; j < ZD; ++j) Lm[lane * ZD + j] = (j == lane) ? sqrtf(S0d[lane]) : 0.f;
  __syncthreads();

  float lp_acc = 0.f;  // lane 0

  for (int t = 0; t < TT; ++t) {
    const float m = mask[b * TT + t];

    // ---------- measurement update ----------
    mm32(Sg, Lm, Lm, true, lane);  // Sigma_pred = L L^T
    __syncthreads();

    if (lane < YD) {  // innovation
      float s = 0.f;
      for (int j = 0; j < ZD; ++j) s += sH[lane * ZD + j] * muv[j];
      innov[lane] = y[(b * TT + t) * YD + lane] - s;
    }
    for (int e = lane; e < YD * ZD; e += 32) {  // HP = H * Sigma
      int i = e >> 5, j = e & 31;
      float s = 0.f;
      for (int k = 0; k < ZD; ++k) s += sH[i * ZD + k] * Sg[k * ZD + j];
      HPm[e] = s;
    }
    __syncthreads();
    for (int e = lane; e < YD * YD; e += 32) {  // S = HP H^T + R + jit
      int i = e >> 4, l = e & 15;
      float s = 0.f;
      for (int j = 0; j < ZD; ++j) s += HPm[i * ZD + j] * sH[l * ZD + j];
      if (i == l) s += sRd[i] + JITF;
      Ssm[e] = s;
    }
    __syncthreads();
    // chol16(S) in place
    for (int k = 0; k < YD; ++k) {
      if (lane == k) Ssm[k * YD + k] = sqrtf(Ssm[k * YD + k]);
      __syncthreads();
      float dk = Ssm[k * YD + k];
      if (lane > k && lane < YD) Ssm[lane * YD + k] /= dk;
      __syncthreads();
      if (lane > k && lane < YD) {
        float lik = Ssm[lane * YD + k];
        for (int j = k + 1; j <= lane; ++j) Ssm[lane * YD + j] -= lik * Ssm[j * YD + k];
      }
      __syncthreads();
    }
    // Solve S X = HP; lane owns column `lane` of X (16x32). Kg(z,y) = X[y][z].
    {
      float z[YD];
      #pragma unroll
      for (int i = 0; i < YD; ++i) {
        float s = HPm[i * ZD + lane];
        #pragma unroll
        for (int j = 0; j < YD; ++j)
          if (j < i) s -= Ssm[i * YD + j] * z[j];
        z[i] = s / Ssm[i * YD + i];
      }
      #pragma unroll
      for (int i = YD - 1; i >= 0; --i) {
        float s = z[i];
        #pragma unroll
        for (int j = 0; j < YD; ++j)
          if (j > i) s -= Ssm[j * YD + i] * z[j];
        z[i] = s / Ssm[i * YD + i];
      }
      #pragma unroll
      for (int i = 0; i < YD; ++i) Xs[i * ZD + lane] = z[i];
    }
    __syncthreads();
    {  // mu_u = mu + Kg innov
      float s = 0.f;
      for (int yv = 0; yv < YD; ++yv) s += Xs[yv * ZD + lane] * innov[yv];
      muu[lane] = muv[lane] + s;
    }
    for (int j = 0; j < ZD; ++j) {  // ImKH row `lane`
      float s = (j == lane) ? 1.f : 0.f;
      for (int yv = 0; yv < YD; ++yv) s -= Xs[yv * ZD + lane] * sH[yv * ZD + j];
      Am[lane * ZD + j] = s;
    }
    __syncthreads();
    mm32(T1, Am, Sg, false, lane);  // ImKH * Sigma
    __syncthreads();
    mm32(T2, T1, Am, true, lane);   // ... * ImKH^T
    __syncthreads();
    for (int e = lane; e < ZD * ZD; e += 32) {  // + Kg R Kg^T + jit I
      int i = e >> 5, j = e & 31;
      float s = 0.f;
      for (int yv = 0; yv < YD; ++yv) s += Xs[yv * ZD + i] * sRd[yv] * Xs[yv * ZD + j];
      T2[e] += s + ((i == j) ? JITF : 0.f);
    }
    __syncthreads();
    chol32(T2, lane);  // L_u

    if (lane == 0) {  // log prob
      float w[YD];
      float qs = 0.f, ldet = 0.f;
      #pragma unroll
      for (int i = 0; i < YD; ++i) {
        float s = innov[i];
        #pragma unroll
        for (int j = 0; j < YD; ++j)
          if (j < i) s -= Ssm[i * YD + j] * w[j];
        w[i] = s / Ssm[i * YD + i];
        qs += w[i] * w[i];
        ldet += logf(Ssm[i * YD + i]);
      }
      lp_acc += m * (-0.5f * (qs + (float)YD * LOG2PI) - ldet);
    }

    // mask blend + commit state
    float mu_new = m * muu[lane] + (1.f - m) * muv[lane];
    __syncthreads();
    muv[lane] = mu_new;
    for (int j = 0; j < ZD; ++j) {
      float lu = (j <= lane) ? T2[lane * ZD + j] : 0.f;
      Lm[lane * ZD + j] = m * lu + (1.f - m) * Lm[lane * ZD + j];
    }
    __syncthreads();

    // outputs: mus (T,B,ZD), Ls (T,B,ZD,ZD)
    out_mus[((size_t)t * BB + b) * ZD + lane] = muv[lane];
    for (int e = lane; e < ZD * ZD; e += 32)
      out_Ls[((size_t)t * BB + b) * (ZD * ZD) + e] = Lm[e];

    // ---------- Euler substeps ----------
    for (int s = 0; s < nsub; ++s) {
      __syncthreads();
      mm32(Sg, Lm, Lm, true, lane);  // Sigma = L L^T
      for (int u = lane; u < HIDN; u += 32) {
        float acc = sb1[u];
        for (int j = 0; j < ZD; ++j) acc += muv[j] * sW1[j * HIDN + u];
        hidv[u] = tanhf(acc);
      }
      __syncthreads();
      if (lane < KMIX) {
        float acc = sb2[lane];
        for (int u = 0; u < HIDN; ++u) acc += hidv[u] * sW2[u * KMIX + lane];
        lgv[lane] = acc;
      }
      __syncthreads();
      {
        float mx = lgv[0];
        for (int k2 = 1; k2 < KMIX; ++k2) mx = fmaxf(mx, lgv[k2]);
        float den = 0.f;
        for (int k2 = 0; k2 < KMIX; ++k2) den += expf(lgv[k2] - mx);
        if (lane < KMIX) alph[lane] = expf(lgv[lane] - mx) / den;
      }
      __syncthreads();
      for (int e = lane; e < ZD * ZD; e += 32) {  // A = sum_k a_k A_base[k]
        float acc = 0.f;
        for (int k2 = 0; k2 < KMIX; ++k2) acc += alph[k2] * Ab[k2 * ZD * ZD + e];
        Am[e] = acc;
      }
      __syncthreads();
      {  // mu2 = mu + h A mu
        float acc = 0.f;
        for (int j = 0; j < ZD; ++j) acc += Am[lane * ZD + j] * muv[j];
        muu[lane] = muv[lane] + h * acc;
      }
      mm32(T1, Am, Sg, false, lane);  // A * Sigma  (Sigma A^T = (A Sigma)^T)
      __syncthreads();
      for (int e = lane; e < ZD * ZD; e += 32) {
        int i = e >> 5, j = e & 31;
        float v = Sg[e] + h * (T1[i * ZD + j] + T1[j * ZD + i]);
        if (i == j) v += h * sQd[i] + JITF;
        T2[e] = v;
      }
      __syncthreads();
      chol32(T2, lane);
      muv[lane] = muu[lane];
      for (int j = 0; j < ZD; ++j)
        Lm[lane * ZD + j] = (j <= lane) ? T2[lane * ZD + j] : 0.f;
      __syncthreads();
    }
  }
  if (lane == 0) out_lp[b] = lp_acc;
}

extern "C" void kernel_launch(void* const* d_in, const int* in_sizes, int n_in,
                              void* d_out, int out_size, void* d_ws, size_t ws_size,
                              hipStream_t stream) {
  const float* y = (const float*)d_in[0];
  const float* mask = (const float*)d_in[1];
  const float* times = (const float*)d_in[2];
  const float* mu0 = (const float*)d_in[3];
  const float* S0d = (const float*)d_in[4];
  const float* H = (const float*)d_in[5];
  const float* Rd = (const float*)d_in[6];
  const float* Ab = (const float*)d_in[7];
  const float* W1 = (const float*)d_in[8];
  const float* b1 = (const float*)d_in[9];
  const float* W2 = (const float*)d_in[10];
  const float* b2 = (const float*)d_in[11];
  const float* Qd = (const float*)d_in[12];
  const int* nsub = (const int*)d_in[13];

  float* out = (float*)d_out;
  float* out_mus = out;                                        // T*B*ZD
  float* out_Ls = out + (size_t)TT * BB * ZD;                  // T*B*ZD*ZD
  float* out_lp = out + (size_t)TT * BB * ZD + (size_t)TT * BB * ZD * ZD;  // B

  kf_scan_kernel<<<BB, 32, 0, stream>>>(y, mask, times, mu0, S0d, H, Rd, Ab, W1, b1, W2, b2,
                                        Qd, nsub, out_mus, out_Ls, out_lp);
}